// SpatialTemporalGridEncoder_44212393345203
// MI455X (gfx1250) — compile-verified
//
#include <hip/hip_runtime.h>
#include <math.h>

// Clang ext-vector types so __builtin_nontemporal_{load,store} accept them.
typedef float v2f __attribute__((ext_vector_type(2)));
typedef float v4f __attribute__((ext_vector_type(4)));

#define NLVL 16

// ---------------------------------------------------------------------------
// Compile-time replication of the reference's _level_grids():
//   res  = ceil(16 * scale^l); grid = res + 1
//   params = min(2^19, prod(grid)) rounded up to multiple of 8
//   use_hash = prod(grid) > params
// Spatial: 3 dims, scale 1.447 (cubic grid). Temporal: dims [g,g,g,g4],
// g4 from scale 1.15. All ceil() arguments are >= 0.01 away from integers,
// so double repeated-multiply is safe vs numpy's pow().
// ---------------------------------------------------------------------------
struct LvlTab {
  int      sg[NLVL];               // spatial grid size (same for x,y,z)
  int      tg4[NLVL];              // temporal 4th-dim grid size
  unsigned soff[NLVL], ssize[NLVL];
  unsigned toff[NLVL], tsize[NLVL];
  int      shash[NLVL], thash[NLVL];
};

constexpr long iceil_pos(double v) { long i = (long)v; return (v > (double)i) ? i + 1 : i; }

constexpr LvlTab make_tab() {
  LvlTab T{};
  double sc3 = 1.0, sc4 = 1.0;
  unsigned so = 0, to = 0;
  for (int l = 0; l < NLVL; ++l) {
    long g  = iceil_pos(16.0 * sc3) + 1;
    long g4 = iceil_pos(16.0 * sc4) + 1;
    long ps = g * g * g;                               // spatial prod(grid)
    long sp = ps < 524288 ? ps : 524288; sp = (sp + 7) / 8 * 8;
    long pt = ps * g4;                                 // temporal prod(grid)
    long tp = pt < 524288 ? pt : 524288; tp = (tp + 7) / 8 * 8;
    T.sg[l] = (int)g;  T.tg4[l] = (int)g4;
    T.soff[l] = so; T.ssize[l] = (unsigned)sp; T.shash[l] = (ps > sp);
    T.toff[l] = to; T.tsize[l] = (unsigned)tp; T.thash[l] = (pt > tp);
    so += (unsigned)sp; to += (unsigned)tp;
    sc3 *= 1.447; sc4 *= 1.15;
  }
  return T;
}

constexpr LvlTab TAB = make_tab();

__global__ void __launch_bounds__(256)
st_grid_encode(const float* __restrict__ inp,   // [B,4]
               const float* __restrict__ semb,  // [soff_end, 2]
               const float* __restrict__ temb,  // [toff_end, 2]
               const float* __restrict__ memb,  // [128^3]
               float* __restrict__ out,         // [B,32]
               int B)
{
  const int b = blockIdx.x * blockDim.x + threadIdx.x;
  if (b >= B) return;

  // Streaming input: NT load (don't pollute L2 — tables must stay resident).
  const v4f x = __builtin_nontemporal_load((const v4f*)inp + b);

  // ------------------- mask: dense trilinear on 128^3, C=1 -----------------
  float mraw;
  {
    const float px = x.x * 127.f, py = x.y * 127.f, pz = x.z * 127.f;
    const float fx = floorf(px), fy = floorf(py), fz = floorf(pz);
    const int ix = (int)fx, iy = (int)fy, iz = (int)fz;
    const float ax = px - fx, ay = py - fy, az = pz - fz;
    float acc = 0.f;
#pragma unroll
    for (int c = 0; c < 8; ++c) {
      const int bx = c & 1, by = (c >> 1) & 1, bz = (c >> 2) & 1;
      const int gx = min(ix + bx, 127), gy = min(iy + by, 127), gz = min(iz + bz, 127);
      const float w = (bx ? ax : 1.f - ax) * (by ? ay : 1.f - ay) * (bz ? az : 1.f - az);
      acc += w * memb[gx + (gy << 7) + (gz << 14)];
    }
    mraw = acc;
  }
  const float m  = 1.f / (1.f + expf(-mraw));   // sigmoid
  const float m1 = 1.f - m;

  float* const orow = out + (size_t)b * 32;

  // --------------------------- 16 levels -----------------------------------
  // Keep the level loop rolled: per-level constants come from scalar loads of
  // the constexpr table (loop index is wave-uniform), I-cache stays small.
#pragma unroll 1
  for (int l = 0; l < NLVL; ++l) {
    const int g   = TAB.sg[l];
    const int g4  = TAB.tg4[l];
    const float gm1 = (float)(g - 1);

    const float px = x.x * gm1, py = x.y * gm1, pz = x.z * gm1;
    const float fx = floorf(px), fy = floorf(py), fz = floorf(pz);
    const int ix = (int)fx, iy = (int)fy, iz = (int)fz;
    const float ax = px - fx, ay = py - fy, az = pz - fz;

    const float pw = x.w * (float)(g4 - 1);
    const float fw = floorf(pw);
    const int   iw = (int)fw;
    const float aw = pw - fw;
    const int   iw1 = min(iw + 1, g4 - 1);

    const v2f* __restrict__ sp = (const v2f*)semb + TAB.soff[l];
    const v2f* __restrict__ tp = (const v2f*)temb + TAB.toff[l];
    const bool sh = TAB.shash[l] != 0;
    const bool th = TAB.thash[l] != 0;
    const unsigned smask = TAB.ssize[l] - 1u;   // 2^19-1 when hashed
    const unsigned tmask = TAB.tsize[l] - 1u;
    const unsigned gu = (unsigned)g;
    const unsigned g3 = gu * gu * gu;           // only used on non-hashed t levels

    // 4th-dim contributions (shared across the 8 xyz corners).
    const unsigned hw0 = (unsigned)iw  * 3674653429u;
    const unsigned hw1 = (unsigned)iw1 * 3674653429u;
    const unsigned lw0 = (unsigned)iw  * g3;
    const unsigned lw1 = (unsigned)iw1 * g3;
    const float w_lo = 1.f - aw, w_hi = aw;

    float s0 = 0.f, s1 = 0.f, t0 = 0.f, t1 = 0.f;
#pragma unroll
    for (int c = 0; c < 8; ++c) {
      const int bx = c & 1, by = (c >> 1) & 1, bz = (c >> 2) & 1;
      const unsigned gx = (unsigned)min(ix + bx, g - 1);
      const unsigned gy = (unsigned)min(iy + by, g - 1);
      const unsigned gz = (unsigned)min(iz + bz, g - 1);
      const float w = (bx ? ax : 1.f - ax) * (by ? ay : 1.f - ay) * (bz ? az : 1.f - az);

      // Spatial and temporal share xyz grid + primes: hash/linear base once.
      const unsigned h   = gx ^ gy * 2654435761u ^ gz * 805459861u;
      const unsigned lin = gx + gu * (gy + gu * gz);

      const unsigned sidx = sh ? (h & smask) : lin;
      const v2f sv = sp[sidx];
      s0 += w * sv.x;
      s1 += w * sv.y;

      const unsigned ti0 = th ? ((h ^ hw0) & tmask) : (lin + lw0);
      const unsigned ti1 = th ? ((h ^ hw1) & tmask) : (lin + lw1);
      const v2f tv0 = tp[ti0];
      const v2f tv1 = tp[ti1];
      const float wlo = w * w_lo, whi = w * w_hi;
      t0 += wlo * tv0.x + whi * tv1.x;
      t1 += wlo * tv0.y + whi * tv1.y;
    }

    // out = m*s + (1-m)*t ; streaming NT store (8B, coalesced per 32-row span)
    v2f o;
    o.x = m * s0 + m1 * t0;
    o.y = m * s1 + m1 * t1;
    __builtin_nontemporal_store(o, (v2f*)(orow + 2 * l));
  }
}

extern "C" void kernel_launch(void* const* d_in, const int* in_sizes, int n_in,
                              void* d_out, int out_size, void* d_ws, size_t ws_size,
                              hipStream_t stream) {
  (void)n_in; (void)out_size; (void)d_ws; (void)ws_size;
  const float* inp  = (const float*)d_in[0];
  const float* semb = (const float*)d_in[1];
  const float* temb = (const float*)d_in[2];
  const float* memb = (const float*)d_in[3];
  float* out = (float*)d_out;

  const int B = in_sizes[0] / 4;            // inputs are [B,4]
  const int block = 256;                    // 8 wave32s per block
  const int grid  = (B + block - 1) / block;
  hipLaunchKernelGGL(st_grid_encode, dim3(grid), dim3(block), 0, stream,
                     inp, semb, temb, memb, out, B);
}